// KFDeepLearningModel_51350628991202
// MI455X (gfx1250) — compile-verified
//
#include <hip/hip_runtime.h>
#include <hip/hip_bf16.h>

typedef __attribute__((ext_vector_type(2))) float v2f;
typedef __attribute__((ext_vector_type(8))) float v8f;

#define T_STEPS 1024
#define KDIM    2048          // 2*T flattened (t-major, obs component inner)
#define NSTEPS  (KDIM / 4)    // 512 WMMA K-steps of 4
#define BROWS   4096

// ---------------- small matrix helpers (row-major) ----------------
__device__ inline void mm44(const float* A, const float* B, float* C) {
#pragma unroll
    for (int i = 0; i < 4; ++i)
#pragma unroll
        for (int j = 0; j < 4; ++j) {
            float s = A[i * 4 + 0] * B[0 * 4 + j];
            s += A[i * 4 + 1] * B[1 * 4 + j];
            s += A[i * 4 + 2] * B[2 * 4 + j];
            s += A[i * 4 + 3] * B[3 * 4 + j];
            C[i * 4 + j] = s;
        }
}

// Build A_t = (I - K H) F from K (4x2). With F=[[1,0,1,0],[0,1,0,1],[0,0,1,0],[0,0,0,1]],
// ((I-KH)F)[i][j] = F[i][j] - K[i][0]*F[0][j] - K[i][1]*F[1][j].
__device__ inline void buildA(const float* K, float* A) {
    const float F[16] = {1, 0, 1, 0, 0, 1, 0, 1, 0, 0, 1, 0, 0, 0, 0, 1};
#pragma unroll
    for (int i = 0; i < 4; ++i) {
        float k0 = K[i * 2 + 0], k1 = K[i * 2 + 1];
        A[i * 4 + 0] = F[i * 4 + 0] - k0;
        A[i * 4 + 1] = F[i * 4 + 1] - k1;
        A[i * 4 + 2] = F[i * 4 + 2] - k0;
        A[i * 4 + 3] = F[i * 4 + 3] - k1;
    }
}

// ---------------- kernel 1: batch-independent weight precompute ----------------
// ws layout (floats): wsB[NSTEPS*64] = B fragments; wsK[T_STEPS*8] = Kalman gains.
__global__ __launch_bounds__(256) void kf_precompute(const float* __restrict__ Qlog,
                                                     const float* __restrict__ Rlog,
                                                     float* __restrict__ wsB,
                                                     float* __restrict__ wsK) {
    __shared__ float CP[256][16];   // per-chunk products A_{4j+3}..A_{4j}
    __shared__ float SC[256][16];   // exclusive suffix of chunk products
    __shared__ float FullProd[16];  // A_{1023}..A_0
    const int tid = threadIdx.x;

    // ---- phase 1: forward Riccati recursion (serial, batch-independent) ----
    if (tid == 0) {
        float Q[16], R[4], P[16];
#pragma unroll
        for (int i = 0; i < 16; ++i)
            Q[i] = __expf(Qlog[i]) + ((i % 5 == 0) ? 1e-6f : 0.0f);
        R[0] = __expf(Rlog[0]) + 1e-6f;
        R[1] = __expf(Rlog[1]);
        R[2] = __expf(Rlog[2]);
        R[3] = __expf(Rlog[3]) + 1e-6f;
#pragma unroll
        for (int i = 0; i < 16; ++i) P[i] = 0.0f;
        P[0] = P[5] = P[10] = P[15] = 1000.0f;

        for (int t = 0; t < T_STEPS; ++t) {
            // P = F P F^T + Q  (F structure: row0+=row2, row1+=row3, then col0+=col2, col1+=col3)
            float Tm[16];
#pragma unroll
            for (int j = 0; j < 4; ++j) {
                Tm[0 * 4 + j] = P[0 * 4 + j] + P[2 * 4 + j];
                Tm[1 * 4 + j] = P[1 * 4 + j] + P[3 * 4 + j];
                Tm[2 * 4 + j] = P[2 * 4 + j];
                Tm[3 * 4 + j] = P[3 * 4 + j];
            }
#pragma unroll
            for (int i = 0; i < 4; ++i) {
                P[i * 4 + 0] = Tm[i * 4 + 0] + Tm[i * 4 + 2] + Q[i * 4 + 0];
                P[i * 4 + 1] = Tm[i * 4 + 1] + Tm[i * 4 + 3] + Q[i * 4 + 1];
                P[i * 4 + 2] = Tm[i * 4 + 2] + Q[i * 4 + 2];
                P[i * 4 + 3] = Tm[i * 4 + 3] + Q[i * 4 + 3];
            }
            // S = P[0:2,0:2] + R ; 2x2 inverse
            float S00 = P[0] + R[0], S01 = P[1] + R[1];
            float S10 = P[4] + R[2], S11 = P[5] + R[3];
            float inv = 1.0f / (S00 * S11 - S01 * S10);
            float i00 = S11 * inv, i01 = -S01 * inv, i10 = -S10 * inv, i11 = S00 * inv;
            // K = P H^T S^{-1} = P[:,0:2] * S^{-1}
            float K[8];
#pragma unroll
            for (int i = 0; i < 4; ++i) {
                K[i * 2 + 0] = P[i * 4 + 0] * i00 + P[i * 4 + 1] * i10;
                K[i * 2 + 1] = P[i * 4 + 0] * i01 + P[i * 4 + 1] * i11;
                wsK[t * 8 + i * 2 + 0] = K[i * 2 + 0];
                wsK[t * 8 + i * 2 + 1] = K[i * 2 + 1];
            }
            // P = (I - K H) P = P - K * P[0:2,:]
            float r0[4], r1[4];
#pragma unroll
            for (int j = 0; j < 4; ++j) { r0[j] = P[0 * 4 + j]; r1[j] = P[1 * 4 + j]; }
#pragma unroll
            for (int i = 0; i < 4; ++i)
#pragma unroll
                for (int j = 0; j < 4; ++j)
                    P[i * 4 + j] -= K[i * 2 + 0] * r0[j] + K[i * 2 + 1] * r1[j];
        }
    }
    __syncthreads();

    // ---- phase 2: per-chunk product (parallel, 4 steps/thread) ----
    {
        float M[16], A[16], Tmp[16];
        buildA(&wsK[(4 * tid) * 8], M);
#pragma unroll
        for (int u = 1; u < 4; ++u) {
            buildA(&wsK[(4 * tid + u) * 8], A);
            mm44(A, M, Tmp);  // later steps multiply on the left
#pragma unroll
            for (int i = 0; i < 16; ++i) M[i] = Tmp[i];
        }
#pragma unroll
        for (int i = 0; i < 16; ++i) CP[tid][i] = M[i];
    }
    __syncthreads();

    // ---- phase 3: exclusive suffix scan of chunk products (serial over 256 chunks) ----
    if (tid == 0) {
        float acc[16] = {1, 0, 0, 0, 0, 1, 0, 0, 0, 0, 1, 0, 0, 0, 0, 1};
        for (int j = 255; j >= 0; --j) {
#pragma unroll
            for (int i = 0; i < 16; ++i) SC[j][i] = acc[i];
            float t2[16];
            mm44(acc, CP[j], t2);
#pragma unroll
            for (int i = 0; i < 16; ++i) acc[i] = t2[i];
        }
#pragma unroll
        for (int i = 0; i < 16; ++i) FullProd[i] = acc[i];  // A_{1023}..A_0
    }
    __syncthreads();

    // ---- phase 4: W_t = S_t K_t, fold output map C, write WMMA B-fragments ----
    {
        float S[16];
#pragma unroll
        for (int i = 0; i < 16; ++i) S[i] = SC[tid][i];
        for (int t = 4 * tid + 3; t >= 4 * tid; --t) {
            float K[8];
#pragma unroll
            for (int i = 0; i < 8; ++i) K[i] = wsK[t * 8 + i];
            float W[8];
#pragma unroll
            for (int i = 0; i < 4; ++i)
#pragma unroll
                for (int c = 0; c < 2; ++c)
                    W[i * 2 + c] = S[i * 4 + 0] * K[0 * 2 + c] + S[i * 4 + 1] * K[1 * 2 + c] +
                                   S[i * 4 + 2] * K[2 * 2 + c] + S[i * 4 + 3] * K[3 * 2 + c];
            if (t == 0) {  // x0 = E z_0 term: W_0 += (A_{1023}..A_0)[:,0:2]
#pragma unroll
                for (int i = 0; i < 4; ++i) {
                    W[i * 2 + 0] += FullProd[i * 4 + 0];
                    W[i * 2 + 1] += FullProd[i * 4 + 1];
                }
            }
            // G (6x2): out row 2(k-1)+r = X[r] + k*X[r+2], k=1..3
            float G[12];
#pragma unroll
            for (int kp = 1; kp <= 3; ++kp)
#pragma unroll
                for (int c = 0; c < 2; ++c) {
                    G[(2 * (kp - 1) + 0) * 2 + c] = W[0 * 2 + c] + (float)kp * W[2 * 2 + c];
                    G[(2 * (kp - 1) + 1) * 2 + c] = W[1 * 2 + c] + (float)kp * W[3 * 2 + c];
                }
            // B-fragment layout for V_WMMA_F32_16X16X4_F32 (K=4 x N=16):
            //   p = k%4: p0 -> lanes 0..15 v0 ; p1 -> lanes 0..15 v1
            //            p2 -> lanes 16..31 v0 ; p3 -> lanes 16..31 v1
            //   stored lane-major pairs: wsB[s*64 + lane*2 + v]
#pragma unroll
            for (int c = 0; c < 2; ++c) {
                int k = 2 * t + c;
                int s = k >> 2;
                int p = k & 3;
                int laneoff = (p >= 2) ? 16 : 0;
                int v = p & 1;
#pragma unroll
                for (int n = 0; n < 16; ++n)
                    wsB[s * 64 + (laneoff + n) * 2 + v] = (n < 6) ? G[n * 2 + c] : 0.0f;
            }
            if (t > 4 * tid) {  // S_{t-1} = S_t * A_t
                float A[16], Tmp[16];
                buildA(K, A);
                mm44(S, A, Tmp);
#pragma unroll
                for (int i = 0; i < 16; ++i) S[i] = Tmp[i];
            }
        }
    }
}

// ---------------- kernel 2: bandwidth-bound WMMA GEMM ----------------
// out(4096x6) = hist(4096x2048) @ Wbig(2048x16-padded)
__global__ __launch_bounds__(256) void kf_gemm(const float* __restrict__ hist,
                                               const float* __restrict__ wsB,
                                               float* __restrict__ out) {
    __shared__ float red[8][256];
    const int tid = threadIdx.x;
    const int lane = tid & 31;
    const int w = tid >> 5;
    const int rowBase = blockIdx.x * 16;
    const int lr = lane & 15;
    const int half = lane >> 4;

    // A fragment (16x4 f32): lanes 0-15 v0/v1 = K+0/K+1 ; lanes 16-31 v0/v1 = K+2/K+3
    const float* arow = hist + (size_t)(rowBase + lr) * KDIM + half * 2;

    v8f acc = {};
    const int s0 = w * (NSTEPS / 8);
#pragma unroll 4
    for (int s = s0; s < s0 + NSTEPS / 8; ++s) {
        v2f a = *(const v2f*)(arow + s * 4);
        v2f b = *(const v2f*)(wsB + s * 64 + lane * 2);
        acc = __builtin_amdgcn_wmma_f32_16x16x4_f32(false, a, false, b, (short)0, acc,
                                                    false, false);
    }

    // reduce 8 partial 16x16 tiles through LDS
#pragma unroll
    for (int e = 0; e < 8; ++e) red[w][lane * 8 + e] = acc[e];
    __syncthreads();
    float sum = 0.0f;
#pragma unroll
    for (int ww = 0; ww < 8; ++ww) sum += red[ww][tid];

    // C/D layout: VGPR e, lanes 0-15 -> M=e, N=lane ; lanes 16-31 -> M=e+8, N=lane-16
    const int clane = tid >> 3;
    const int e = tid & 7;
    const int M = e + ((clane >= 16) ? 8 : 0);
    const int N = clane & 15;
    if (N < 6) out[(size_t)(rowBase + M) * 6 + N] = sum;
}

extern "C" void kernel_launch(void* const* d_in, const int* in_sizes, int n_in,
                              void* d_out, int out_size, void* d_ws, size_t ws_size,
                              hipStream_t stream) {
    const float* hist = (const float*)d_in[0];  // (4096,1024,2)
    const float* Qlog = (const float*)d_in[1];  // (4,4)
    const float* Rlog = (const float*)d_in[2];  // (2,2)
    float* out = (float*)d_out;                 // (4096,3,2)
    float* ws = (float*)d_ws;
    float* wsB = ws;                 // NSTEPS*64 = 32768 floats (128 KB)
    float* wsK = ws + NSTEPS * 64;   // T_STEPS*8 = 8192 floats (32 KB)

    hipLaunchKernelGGL(kf_precompute, dim3(1), dim3(256), 0, stream, Qlog, Rlog, wsB, wsK);
    hipLaunchKernelGGL(kf_gemm, dim3(BROWS / 16), dim3(256), 0, stream, hist, wsB, out);
}